// SpectralConvS2_13176959664731
// MI455X (gfx1250) — compile-verified
//
#include <hip/hip_runtime.h>

// ---------------- problem dims (fixed by the reference) ----------------
#define CIN_   256
#define COUT_  256
#define NLAT_  256
#define NLON_  512
#define LMAX_  256
#define MMAX_  256

typedef float v2f __attribute__((ext_vector_type(2)));
typedef float v8f __attribute__((ext_vector_type(8)));

#define LDS_PITCH 36   // 36-float row pitch: 16B-aligned rows, and the 32 lanes'
                       // b64 fragment reads tile all 64 LDS banks exactly once.

// =======================================================================
// Generic batched, strided fp32 GEMM on v_wmma_f32_16x16x4_f32.
//   C[z](i,j) (+)= alpha * sum_k A[z](i,k) * B[z](k,j)
// addresses (elements):
//   A: offA + z*batA + i*sAi + k*sAk
//   B: offB + z*batB + k*sBk + j*sBj
//   C: offC + z*batC + i*sCi + j*sCj
// Requires M%64==0, N%64==0, K%32==0 (true for every call below).
// Block = 128 threads (4 wave32 waves), tile 64x64, 32x32 per wave.
// Wave-uniform float4 fast paths for unit-stride operands (EXEC stays all-1).
// =======================================================================
__global__ __launch_bounds__(128) void gemm_f32_wmma(
    const float* __restrict__ A, const float* __restrict__ Bm, float* __restrict__ C,
    int K,
    long long offA, long long batA, long long sAi, long long sAk,
    long long offB, long long batB, long long sBk, long long sBj,
    long long offC, long long batC, long long sCi, long long sCj,
    float alpha, int accum)
{
    __shared__ float As[64 * LDS_PITCH];
    __shared__ float Bs[64 * LDS_PITCH];

    const int tid  = threadIdx.x;
    const int lane = tid & 31;
    const int wave = tid >> 5;
    const int wr   = (wave >> 1) * 32;   // wave tile origin in M
    const int wc   = (wave & 1) * 32;    // wave tile origin in N
    const int lh   = lane & 15;
    const int hi   = lane >> 4;          // lane half selects K pair / M+8

    const long long i0 = (long long)blockIdx.y * 64;
    const long long j0 = (long long)blockIdx.x * 64;

    const float* Ab = A  + offA + (long long)blockIdx.z * batA;
    const float* Bb = Bm + offB + (long long)blockIdx.z * batB;
    float*       Cb = C  + offC + (long long)blockIdx.z * batC;

    // Wave-uniform (SGPR) fast-path predicates; 16B global alignment guaranteed
    // when the non-unit strides/offsets are multiples of 4 floats.
    const bool vecA  = (sAk == 1) && !((sAi | offA | batA) & 3);
    const bool vecBk = (sBk == 1) && !((sBj | offB | batB) & 3);
    const bool vecBj = (sBj == 1) && !((sBk | offB | batB) & 3);
    const bool vecC  = (sCi == 1) && !accum && !((sCj | offC | batC) & 3);

    const v8f vzero = {0.f, 0.f, 0.f, 0.f, 0.f, 0.f, 0.f, 0.f};
    v8f acc[2][2];
    acc[0][0] = vzero; acc[0][1] = vzero; acc[1][0] = vzero; acc[1][1] = vzero;

    for (int kb = 0; kb < K; kb += 32) {
        // ---- stage A tile 64x32 into LDS as As[i][k] ----
        if (vecA) {
            #pragma unroll
            for (int e = 0; e < 4; ++e) {              // 512 float4 / 128 thr
                int idx = e * 128 + tid;
                int i = idx >> 3, k4 = (idx & 7) * 4;
                float4 v = *(const float4*)&Ab[(i0 + i) * sAi + (kb + k4)];
                *(float4*)&As[i * LDS_PITCH + k4] = v; // b128, 16B aligned
            }
        } else {
            #pragma unroll
            for (int e = 0; e < 16; ++e) {
                int idx = e * 128 + tid;
                int i = idx >> 5, k = idx & 31;
                As[i * LDS_PITCH + k] =
                    Ab[(i0 + i) * sAi + (long long)(kb + k) * sAk];
            }
        }
        // ---- stage B tile 32x64 into LDS transposed as Bs[j][k] ----
        if (vecBk) {                                    // B k-contiguous
            #pragma unroll
            for (int e = 0; e < 4; ++e) {
                int idx = e * 128 + tid;
                int j = idx >> 3, k4 = (idx & 7) * 4;
                float4 v = *(const float4*)&Bb[(long long)(kb + k4) + (j0 + j) * sBj];
                *(float4*)&Bs[j * LDS_PITCH + k4] = v;
            }
        } else if (vecBj) {                             // B j-contiguous
            #pragma unroll
            for (int e = 0; e < 4; ++e) {
                int idx = e * 128 + tid;
                int k = idx >> 4, j4 = (idx & 15) * 4;
                float4 v = *(const float4*)&Bb[(long long)(kb + k) * sBk + (j0 + j4)];
                Bs[(j4 + 0) * LDS_PITCH + k] = v.x;
                Bs[(j4 + 1) * LDS_PITCH + k] = v.y;
                Bs[(j4 + 2) * LDS_PITCH + k] = v.z;
                Bs[(j4 + 3) * LDS_PITCH + k] = v.w;
            }
        } else {
            #pragma unroll
            for (int e = 0; e < 16; ++e) {
                int idx = e * 128 + tid;
                int k = idx >> 6, j = idx & 63;
                Bs[j * LDS_PITCH + k] =
                    Bb[(long long)(kb + k) * sBk + (j0 + j) * sBj];
            }
        }
        __syncthreads();

        // ---- 8 K-steps of 4: fragments via ds_load_b64, conflict-free ----
        #pragma unroll
        for (int kk = 0; kk < 32; kk += 4) {
            const int ko = kk + hi * 2;          // lanes 16-31 hold K+2,K+3
            v2f a0 = *(const v2f*)&As[(wr + lh)      * LDS_PITCH + ko];
            v2f a1 = *(const v2f*)&As[(wr + 16 + lh) * LDS_PITCH + ko];
            v2f b0 = *(const v2f*)&Bs[(wc + lh)      * LDS_PITCH + ko];
            v2f b1 = *(const v2f*)&Bs[(wc + 16 + lh) * LDS_PITCH + ko];
            acc[0][0] = __builtin_amdgcn_wmma_f32_16x16x4_f32(false, a0, false, b0, (short)0, acc[0][0], false, false);
            acc[0][1] = __builtin_amdgcn_wmma_f32_16x16x4_f32(false, a0, false, b1, (short)0, acc[0][1], false, false);
            acc[1][0] = __builtin_amdgcn_wmma_f32_16x16x4_f32(false, a1, false, b0, (short)0, acc[1][0], false, false);
            acc[1][1] = __builtin_amdgcn_wmma_f32_16x16x4_f32(false, a1, false, b1, (short)0, acc[1][1], false, false);
        }
        __syncthreads();
    }

    // Epilogue. C/D layout: VGPR r -> M=r (lanes 0-15) / M=r+8 (lanes 16-31), N=lane%16.
    if (vecC) {
        // Per lane the 8 accumulator rows are 8 consecutive floats (sCi==1):
        // two b128 stores per fragment.
        #pragma unroll
        for (int mi = 0; mi < 2; ++mi) {
            #pragma unroll
            for (int ni = 0; ni < 2; ++ni) {
                long long row = i0 + wr + mi * 16 + hi * 8;
                long long col = j0 + wc + ni * 16 + lh;
                float* p = &Cb[row + col * sCj];
                float4 v0 = {alpha * acc[mi][ni][0], alpha * acc[mi][ni][1],
                             alpha * acc[mi][ni][2], alpha * acc[mi][ni][3]};
                float4 v1 = {alpha * acc[mi][ni][4], alpha * acc[mi][ni][5],
                             alpha * acc[mi][ni][6], alpha * acc[mi][ni][7]};
                *(float4*)(p)     = v0;
                *(float4*)(p + 4) = v1;
            }
        }
    } else {
        #pragma unroll
        for (int mi = 0; mi < 2; ++mi) {
            #pragma unroll
            for (int ni = 0; ni < 2; ++ni) {
                #pragma unroll
                for (int r = 0; r < 8; ++r) {
                    long long row = i0 + wr + mi * 16 + hi * 8 + r;
                    long long col = j0 + wc + ni * 16 + lh;
                    long long ofs = row * sCi + col * sCj;
                    float v = alpha * acc[mi][ni][r];
                    if (accum) Cb[ofs] += v; else Cb[ofs] = v;
                }
            }
        }
    }
}

// =======================================================================
// Table builders (exact integer mod-512 phase reduction).
// Fre/Fim: forward rfft matrix [k=0..511][m=0..255], norm="forward" (1/N).
// Cr/Ci:   irfft matrix [m=0..255][k=0..511], norm="forward" (no 1/N),
//          Im of DC bin dropped, modes m>=1 doubled, Nyquist absent (zero pad).
// =======================================================================
__global__ void sc_build_tables(float* __restrict__ Fre, float* __restrict__ Fim,
                                float* __restrict__ Cr,  float* __restrict__ Ci)
{
    int gid = blockIdx.x * blockDim.x + threadIdx.x;   // 131072 total
    const float w = 6.28318530717958647692f / 512.0f;
    {
        int k = gid >> 8, m = gid & 255;
        int r = (k * m) & 511;
        float a = w * (float)r;
        Fre[gid] =  cosf(a) * (1.0f / 512.0f);
        Fim[gid] = -sinf(a) * (1.0f / 512.0f);
    }
    {
        int m = gid >> 9, k = gid & 511;
        int r = (m * k) & 511;
        float a = w * (float)r;
        Cr[gid] = (m == 0) ? 1.0f :  2.0f * cosf(a);
        Ci[gid] = (m == 0) ? 0.0f : -2.0f * sinf(a);
    }
}

// Pw[l,m,n] = P[l,m,n] * wquad[n]   (fold quadrature weights into fwd Legendre)
__global__ void sc_build_pw(const float* __restrict__ P, const float* __restrict__ wq,
                            float* __restrict__ Pw)
{
    int gid = blockIdx.x * blockDim.x + threadIdx.x;   // 16,777,216 total
    Pw[gid] = P[gid] * wq[gid & (NLAT_ - 1)];
}

// Residual output: second tuple element is x, copied verbatim (float4).
__global__ void sc_copy4(const float4* __restrict__ src, float4* __restrict__ dst)
{
    long long gid = (long long)blockIdx.x * blockDim.x + threadIdx.x;
    dst[gid] = src[gid];
}

// =======================================================================
extern "C" void kernel_launch(void* const* d_in, const int* in_sizes, int n_in,
                              void* d_out, int out_size, void* d_ws, size_t ws_size,
                              hipStream_t stream)
{
    const float* x  = (const float*)d_in[0];   // [1,256,256,512]
    const float* wt = (const float*)d_in[1];   // [256,256,256,2]
    const float* P  = (const float*)d_in[2];   // [256,256,256]
    const float* wq = (const float*)d_in[3];   // [256]
    float* out = (float*)d_out;                // [out(33.5M) | residual(33.5M)]

    // ---------------- workspace layout (floats) ----------------
    float* ws   = (float*)d_ws;
    const long long SZF = 131072;              // 512*256
    float* Fre  = ws;                          // fwd DFT (re)
    float* Fim  = Fre + SZF;                   // fwd DFT (im)
    float* Cr   = Fim + SZF;                   // inv DFT (re)
    float* Ci   = Cr  + SZF;                   // inv DFT (im)
    float* Pw   = Ci  + SZF;                   // 16,777,216 : P * wquad
    float* bufA = Pw  + 16777216LL;            // 33,554,432 : complex ping
    float* bufB = bufA + 33554432LL;           // 33,554,432 : complex pong
    const long long H = 16777216LL;            // re/im half offset

    sc_build_tables<<<512, 256, 0, stream>>>(Fre, Fim, Cr, Ci);
    sc_build_pw<<<65536, 256, 0, stream>>>(P, wq, Pw);

    auto gemm = [&](const float* Ax, const float* Bx, float* Cx,
                    int M, int N, int K, int batch,
                    long long oA, long long bA, long long sAi, long long sAk,
                    long long oB, long long bB, long long sBk, long long sBj,
                    long long oC, long long bC, long long sCi, long long sCj,
                    float alpha, int accum) {
        dim3 g(N / 64, M / 64, batch);
        gemm_f32_wmma<<<g, 128, 0, stream>>>(Ax, Bx, Cx, K,
            oA, bA, sAi, sAk, oB, bB, sBk, sBj, oC, bC, sCi, sCj, alpha, accum);
    };

    // Stage 1: forward DFT over longitude (GEMM vs 512x256 Fourier matrix).
    //   xf[m][c][n] = sum_k x[c,n,k] * F[k][m]   (bufA: re | im halves)
    gemm(x, Fre, bufA, 65536, 256, 512, 1,  0,0, 512,1,   0,0, 256,1,   0,0, 1,65536,  1.f, 0);
    gemm(x, Fim, bufA, 65536, 256, 512, 1,  0,0, 512,1,   0,0, 256,1,   H,0, 1,65536,  1.f, 0);

    // Stage 2: forward Legendre, batched over m (quadrature folded into Pw).
    //   coeffs[l][c][m] = sum_n xf[m][c][n] * Pw[l,m,n]
    gemm(bufA, Pw, bufB, 256, 256, 256, 256,  0,65536, 256,1,  0,256, 1,65536,  0,1, 256,65536, 1.f, 0);
    gemm(bufA, Pw, bufB, 256, 256, 256, 256,  H,65536, 256,1,  0,256, 1,65536,  H,1, 256,65536, 1.f, 0);

    // Stage 3: per-degree complex channel mix, batched over l.
    //   y_re = Wre*Cre - Wim*Cim ;  y_im = Wre*Cim + Wim*Cre   (y in bufA)
    gemm(wt, bufB, bufA, 256, 256, 256, 256,  0,2, 131072,512,  0,65536, 256,1,  0,65536, 256,1,  1.f, 0);
    gemm(wt, bufB, bufA, 256, 256, 256, 256,  1,2, 131072,512,  H,65536, 256,1,  0,65536, 256,1, -1.f, 1);
    gemm(wt, bufB, bufA, 256, 256, 256, 256,  0,2, 131072,512,  H,65536, 256,1,  H,65536, 256,1,  1.f, 0);
    gemm(wt, bufB, bufA, 256, 256, 256, 256,  1,2, 131072,512,  0,65536, 256,1,  H,65536, 256,1,  1.f, 1);

    // Stage 4: inverse Legendre, batched over m (unweighted P).
    //   xh[o][n][m] = sum_l y[l][o][m] * P[l,m,n]   (xh in bufB)
    gemm(bufA, P, bufB, 256, 256, 256, 256,  0,1, 256,65536,  0,256, 65536,1,  0,1, 65536,256, 1.f, 0);
    gemm(bufA, P, bufB, 256, 256, 256, 256,  H,1, 256,65536,  0,256, 65536,1,  H,1, 65536,256, 1.f, 0);

    // Stage 5: inverse DFT over longitude, straight into d_out.
    //   out[o,n,k] = sum_m xh_re*Cr[m][k] + xh_im*Ci[m][k]
    gemm(bufB, Cr, out, 65536, 512, 256, 1,  0,0, 256,1,  0,0, 512,1,  0,0, 512,1, 1.f, 0);
    gemm(bufB, Ci, out, 65536, 512, 256, 1,  H,0, 256,1,  0,0, 512,1,  0,0, 512,1, 1.f, 1);

    // Residual (second tuple output) = x.
    sc_copy4<<<32768, 256, 0, stream>>>((const float4*)x, (float4*)(out + 33554432LL));
}